// GVAE_24833500906043
// MI455X (gfx1250) — compile-verified
//
#include <hip/hip_runtime.h>
#include <math.h>
#include <stdint.h>

typedef float v2f __attribute__((ext_vector_type(2)));
typedef float v8f __attribute__((ext_vector_type(8)));

// D = A(16x4) * B(4x16) + C, all f32, wave32.
#define WMMA_F32(a, b, c) \
    __builtin_amdgcn_wmma_f32_16x16x4_f32(false, (a), false, (b), (short)0, (c), false, false)

namespace {
constexpr int B_   = 16;
constexpr int INP  = 8192;
constexpr int OUTP = 8192;
constexpr int M_   = 16;
constexpr int CIN  = 32;
constexpr int COUT = 32;
constexpr int W_   = 16;
constexpr int K_   = W_ * CIN;   // 512; step-2 contraction index K = i*16 + w
constexpr int XS   = 260;        // per-b LDS stride (16*16 + 4): lane banks l16*4, conflict-free
constexpr int TS   = 17;         // tT row stride: step-2 B reads lane-stride 1, conflict-free
constexpr int WR_ELEMS = (K_ / 2) * COUT * 2;   // 16384 floats = 64 KB
}

// Async global->LDS DMA (gfx1250 VGLOBAL op 98). vdst = per-lane LDS byte address,
// vaddr = per-lane 64-bit global address. Tracked by ASYNCcnt.
__device__ __forceinline__ void async_g2l_b128(unsigned lds_off, const float* g) {
    asm volatile("global_load_async_to_lds_b128 %0, %1, off"
                 :: "v"(lds_off), "v"((unsigned long long)(uintptr_t)g)
                 : "memory");
}
#define WAIT_ASYNC_LE16() asm volatile("s_wait_asynccnt 0x10" ::: "memory")
#define WAIT_ASYNC_0()    asm volatile("s_wait_asynccnt 0x0"  ::: "memory")

// One-shot reorder of weights into the exact step-2 WMMA A-operand lane layout:
//   Wr[((kc*2 + hi)*32 + o)*2 + j] = W2[o][K],  K = kc*4 + hi*2 + j = i*16 + w
__global__ __launch_bounds__(256)
void reorder_wts(const float* __restrict__ wts, float* __restrict__ wr)
{
    int t = blockIdx.x * 256 + threadIdx.x;
    if (t >= WR_ELEMS) return;
    int j  = t & 1;
    int o  = (t >> 1) & 31;
    int kh = t >> 6;                 // kc*2 + hi
    int K  = (kh >> 1) * 4 + (kh & 1) * 2 + j;
    int i  = K >> 4;
    int w  = K & 15;
    wr[t] = wts[w * (COUT * CIN) + o * CIN + i];
}

template <bool USE_WR>
__global__ __launch_bounds__(64)
void gvae_fused(const float* __restrict__ x,      // (B, IN_PN, CIN)
                const int*   __restrict__ nid,    // (OUT_PN, M)
                const float* __restrict__ wts,    // (WNUM, COUT*CIN)
                const float* __restrict__ bias,   // (COUT)
                const float* __restrict__ wwg,    // (OUT_PN, M, WNUM)
                const float* __restrict__ wr,     // reordered weights (or null)
                float* __restrict__ out)          // (B, OUT_PN, COUT)
{
    __shared__ __align__(16) float xg[2][B_ * XS]; // double-buffered gather (2 x 16.6 KB)
    __shared__ float tT[K_ * TS];                  // 34.8 KB: tT[i*16+w][b]
    __shared__ float wwl[M_ * W_];                 // masked w_weights
    __shared__ int   nidl[M_];

    const int p    = blockIdx.x;
    const int tid  = threadIdx.x;
    const int lane = tid & 31;
    const int wv   = tid >> 5;     // 0..1
    const int l16  = lane & 15;
    const int hi   = lane >> 4;    // selects K-pair {0,1} vs {2,3} inside A/B tiles

    // Block-uniform neighbor ids -> scalar loads (fully unrolled, constant m).
    int rm[M_];
    #pragma unroll
    for (int m = 0; m < M_; ++m) {
        int nb = nid[p * M_ + m];
        rm[m] = (nb > INP - 1) ? (INP - 1) : nb;   // clamp: masked ww zeroes invalid m anyway
        if (tid == 0) nidl[m] = nb;
    }
    __syncthreads();

    // Kick off BOTH channel-chunk gathers asynchronously (16 async b128 per chunk per thread's
    // (b, quad) slot). Async loads complete in order, so asynccnt<=16 means chunk 0 landed.
    {
        const int gb = tid >> 2;            // b
        const int gq = tid & 3;             // float4 quad within a 16-ch chunk
        const float* g0 = x + (size_t)gb * INP * CIN + gq * 4;
        unsigned l0 = (unsigned)(uintptr_t)&xg[0][gb * XS + gq * 4];
        unsigned l1 = (unsigned)(uintptr_t)&xg[1][gb * XS + gq * 4];
        #pragma unroll
        for (int m = 0; m < M_; ++m)
            async_g2l_b128(l0 + m * 64, g0 + (size_t)rm[m] * CIN);
        #pragma unroll
        for (int m = 0; m < M_; ++m)
            async_g2l_b128(l1 + m * 64, g0 + (size_t)rm[m] * CIN + 16);
    }

    // masked ww while the DMA flies: ww[m][w] = w_weights[p,m,w] * (nid[p,m] < IN_PN)
    #pragma unroll
    for (int t4 = 0; t4 < 4; ++t4) {
        int t = tid + t4 * 64;
        int m = t >> 4;
        float msk = (nidl[m] < INP) ? 1.0f : 0.0f;
        wwl[t] = wwg[(size_t)p * (M_ * W_) + t] * msk;
    }

    // Step 1 on one chunk: tT[(i*16+w)][b] = sum_m xg[b][m][i_local] * ww[m][w]
    // Constant trip count (8) -> full unroll, no exec-masked backedge, deep ds-load pipelining.
    auto step1 = [&](int ic) {
        #pragma unroll
        for (int mi = 0; mi < 8; ++mi) {
            const int ml = wv + 2 * mi;
            v8f acc = {0.f, 0.f, 0.f, 0.f, 0.f, 0.f, 0.f, 0.f};
            #pragma unroll
            for (int kc = 0; kc < 4; ++kc) {
                int k0 = kc * 4 + hi * 2;
                v2f a, bb;
                a.x  = xg[ic][l16 * XS + (k0 + 0) * 16 + ml];
                a.y  = xg[ic][l16 * XS + (k0 + 1) * 16 + ml];
                bb.x = wwl[(k0 + 0) * W_ + l16];
                bb.y = wwl[(k0 + 1) * W_ + l16];
                acc = WMMA_F32(a, bb, acc);
            }
            int i = ic * 16 + ml;
            #pragma unroll
            for (int g = 0; g < 8; ++g)
                tT[(i * 16 + l16) * TS + (g + hi * 8)] = acc[g];
        }
    };

    WAIT_ASYNC_LE16();     // chunk 0 resident (this wave's share)
    __syncthreads();       // ... and every wave's share
    step1(0);              // chunk-1 DMA streams in behind these WMMAs
    WAIT_ASYNC_0();
    __syncthreads();
    step1(1);
    __syncthreads();

    // ---- Step 2: outT[o][b] = sum_K W2[o][K] * tT[K][b]
    v8f acc0 = {0.f, 0.f, 0.f, 0.f, 0.f, 0.f, 0.f, 0.f};
    v8f acc1 = {0.f, 0.f, 0.f, 0.f, 0.f, 0.f, 0.f, 0.f};
    const int o = wv * 16 + l16;        // A-tile row owned by this lane

    #pragma unroll 4
    for (int kc = 0; kc < K_ / 4; kc += 2) {
        v2f a0, a1;
        if constexpr (USE_WR) {
            a0 = ((const v2f*)wr)[(kc * 2 + hi) * 32 + o];        // coalesced b64
            a1 = ((const v2f*)wr)[((kc + 1) * 2 + hi) * 32 + o];
        } else {
            int k0 = kc * 4 + hi * 2;
            int k1 = k0 + 4;
            a0.x = wts[((k0    ) & 15) * (COUT * CIN) + o * CIN + ((k0    ) >> 4)];
            a0.y = wts[((k0 + 1) & 15) * (COUT * CIN) + o * CIN + ((k0 + 1) >> 4)];
            a1.x = wts[((k1    ) & 15) * (COUT * CIN) + o * CIN + ((k1    ) >> 4)];
            a1.y = wts[((k1 + 1) & 15) * (COUT * CIN) + o * CIN + ((k1 + 1) >> 4)];
        }
        int k0 = kc * 4 + hi * 2;
        int k1 = k0 + 4;
        v2f b0, b1;
        b0.x = tT[(k0    ) * TS + l16];
        b0.y = tT[(k0 + 1) * TS + l16];
        b1.x = tT[(k1    ) * TS + l16];
        b1.y = tT[(k1 + 1) * TS + l16];
        acc0 = WMMA_F32(a0, b0, acc0);
        acc1 = WMMA_F32(a1, b1, acc1);
    }

    // D tile: row = o_local = g + hi*8, col = b = l16. Bias + ELU + store.
    #pragma unroll
    for (int g = 0; g < 8; ++g) {
        int oo = wv * 16 + hi * 8 + g;
        float v = acc0[g] + acc1[g] + bias[oo];
        v = (v > 0.f) ? v : (expf(v) - 1.0f);
        out[((size_t)l16 * OUTP + p) * COUT + oo] = v;   // 8 consecutive floats per lane
    }
}

extern "C" void kernel_launch(void* const* d_in, const int* in_sizes, int n_in,
                              void* d_out, int out_size, void* d_ws, size_t ws_size,
                              hipStream_t stream) {
    const float* x    = (const float*)d_in[0];
    const int*   nidp = (const int*)  d_in[1];
    const float* wts  = (const float*)d_in[2];
    const float* bias = (const float*)d_in[3];
    const float* wwg  = (const float*)d_in[4];
    float*       out  = (float*)d_out;

    if (ws_size >= WR_ELEMS * sizeof(float)) {
        float* wr = (float*)d_ws;
        reorder_wts<<<dim3((WR_ELEMS + 255) / 256), dim3(256), 0, stream>>>(wts, wr);
        gvae_fused<true><<<dim3(OUTP), dim3(64), 0, stream>>>(x, nidp, wts, bias, wwg, wr, out);
    } else {
        gvae_fused<false><<<dim3(OUTP), dim3(64), 0, stream>>>(x, nidp, wts, bias, wwg, nullptr, out);
    }

    (void)in_sizes; (void)n_in; (void)out_size;
}